// MaskedBEVAttention_24721831756414
// MI455X (gfx1250) — compile-verified
//
#include <hip/hip_runtime.h>
#include <math.h>

// ---------------------------------------------------------------------------
// MaskedBEVAttention for MI455X (gfx1250, wave32)
//   B=2, Nr=16384, M=64, ADIM=128, HID=512, NUM_HEADS=4, d=32
// Memory-bound: ~2.15 GB of K/V streamed once -> ~93us floor @ 23.3 TB/s.
// Attention = batched GEMV -> VALU (+ wave32 shuffle softmax, prefetch).
// MLP = per-16-token-tile fp32 GEMMs -> V_WMMA_F32_16X16X4_F32.
// Q tile staged with GLOBAL_LOAD_ASYNC_TO_LDS (ASYNCcnt).
// ---------------------------------------------------------------------------

#define B_    2
#define NR    16384
#define M_    64
#define ADIM  128
#define HID   512
#define NH    4
#define DH    32
#define TPB   256
#define TOK   16          // tokens per workgroup

typedef float v2f  __attribute__((ext_vector_type(2)));
typedef float v8f  __attribute__((ext_vector_type(8)));
typedef float f32x4 __attribute__((ext_vector_type(4)));

__device__ __forceinline__ float gelu_exact(float x) {
  // jax.nn.gelu(approximate=False): x * 0.5 * (1 + erf(x/sqrt(2)))
  return 0.5f * x * (1.0f + erff(x * 0.70710678118654752f));
}

__device__ __forceinline__ v8f wmma_f32_4(v2f a, v2f b, v8f c) {
  return __builtin_amdgcn_wmma_f32_16x16x4_f32(
      /*neg_a=*/false, a, /*neg_b=*/false, b,
      /*c_mod=*/(short)0, c, /*reuse_a=*/false, /*reuse_b=*/false);
}

// LayerNorm over ADIM=128 for a 16-token tile held in LDS.
// 256 threads = 16 tokens x 16 threads, each owning 8 channels.
__device__ __forceinline__ void layernorm_tile(
    float (*src)[ADIM], float (*dst)[ADIM],
    const float* __restrict__ g, const float* __restrict__ bta,
    float* red, float* red2, float* mu_s, float* rs_s, int tid) {
  const int tt = tid >> 4;
  const int i  = tid & 15;
  float sm = 0.f, sq = 0.f;
#pragma unroll
  for (int j = 0; j < 8; ++j) {
    float v = src[tt][i * 8 + j];
    sm += v; sq += v * v;
  }
  red[tid] = sm; red2[tid] = sq;
  __syncthreads();
  for (int s = 8; s >= 1; s >>= 1) {
    if ((tid & 15) < s) { red[tid] += red[tid + s]; red2[tid] += red2[tid + s]; }
    __syncthreads();
  }
  if ((tid & 15) == 0) {
    float mu  = red[tid] * (1.0f / ADIM);
    float var = red2[tid] * (1.0f / ADIM) - mu * mu;
    mu_s[tt] = mu;
    rs_s[tt] = rsqrtf(var + 1e-5f);
  }
  __syncthreads();
  const float mu = mu_s[tt], rs = rs_s[tt];
#pragma unroll
  for (int j = 0; j < 8; ++j) {
    int c = i * 8 + j;
    dst[tt][c] = (src[tt][c] - mu) * rs * g[c] + bta[c];
  }
  __syncthreads();
}

__global__ __launch_bounds__(TPB) void bev_block_kernel(
    const float* __restrict__ Q,            // [B,NR,ADIM]
    const float* __restrict__ K,            // [B,NR,M,ADIM]
    const float* __restrict__ V,            // [B,NR,M,ADIM]
    const unsigned char* __restrict__ mask, // [NR,M] (numpy bool, 1 byte)
    const float* __restrict__ ln1g, const float* __restrict__ ln1b,
    const float* __restrict__ ln2g, const float* __restrict__ ln2b,
    const float* __restrict__ W1, const float* __restrict__ bias1,
    const float* __restrict__ W2, const float* __restrict__ bias2,
    float* __restrict__ out) {
  __shared__ float s_x[TOK][ADIM];     // Q -> Q+attn -> LN1(x)   (8 KB)
  __shared__ float s_y[TOK][ADIM];     // LN2(x)                  (8 KB)
  __shared__ float s_h[TOK][HID];      // gelu(y@W1+b1)           (32 KB)
  __shared__ float s_attn[NH][M_];     // per-token attn weights  (1 KB)
  __shared__ float s_red[TPB];         // AV partials / LN scratch
  __shared__ float s_red2[TPB];        // LN scratch
  __shared__ float s_wredA[8];         // per-wave max partials
  __shared__ float s_wredB[8];         // per-wave sum partials
  __shared__ float s_mu[TOK], s_rs[TOK];

  const int tid  = threadIdx.x;
  const int lane = tid & 31;
  const int wave = tid >> 5;
  const long tokBase = (long)blockIdx.x * TOK;

  // ---- stage the 16-token Q tile with async copy (ASYNCcnt path) ----
  // LDS aperture maps generic addr[31:0] -> LDS byte address, so the low
  // 32 bits of a generic LDS pointer are a valid VDST operand.
  {
    const float* gQ = Q + (size_t)tokBase * ADIM;
    for (int i = tid; i < (TOK * ADIM) / 4; i += TPB) {       // 512 x b128
      unsigned     ldsa = (unsigned)(uintptr_t)(&s_x[0][0] + i * 4);
      const float* gp   = gQ + i * 4;
      asm volatile("global_load_async_to_lds_b128 %0, %1, off"
                   :: "v"(ldsa), "v"(gp) : "memory");
    }
    asm volatile("s_wait_asynccnt 0x0" ::: "memory");
  }
  __syncthreads();

  // =================== attention, one token at a time ===================
  for (int t = 0; t < TOK; ++t) {
    const long tok = tokBase + t;
    const int  n   = (int)(tok & (NR - 1));   // pad_mask indexed by n only

    // prefetch this token's V tile: 256 threads x 128B = 32 KB
    __builtin_prefetch(V + (size_t)(tok * M_) * ADIM + tid * 32, 0, 3);

    const int h = tid >> 6;
    const int m = tid & 63;

    // --- logit: one 32-length dot per thread (4 heads x 64 m) ---
    float lg;
    {
      const float* kp = K + ((size_t)(tok * M_ + m)) * ADIM + h * DH;
      float acc = 0.f;
#pragma unroll
      for (int k4 = 0; k4 < DH / 4; ++k4) {
        f32x4 kv = __builtin_nontemporal_load((const f32x4*)kp + k4);  // TH=NT: streamed once
        acc += kv.x * s_x[t][h * DH + k4 * 4 + 0];
        acc += kv.y * s_x[t][h * DH + k4 * 4 + 1];
        acc += kv.z * s_x[t][h * DH + k4 * 4 + 2];
        acc += kv.w * s_x[t][h * DH + k4 * 4 + 3];
      }
      lg = mask[n * M_ + m] ? -INFINITY
                            : acc * 0.17677669529663687f;  // 1/sqrt(32)
    }

    // --- masked softmax over m=64 (2 waves): shuffle + tiny LDS combine ---
    float wmax = lg;
#pragma unroll
    for (int s = 16; s >= 1; s >>= 1)
      wmax = fmaxf(wmax, __shfl_xor(wmax, s, 32));
    if (lane == 0) s_wredA[wave] = wmax;
    __syncthreads();
    const float maxv = fmaxf(s_wredA[h * 2], s_wredA[h * 2 + 1]);
    const float e    = expf(lg - maxv);        // exp(-inf)=0 for masked
    float wsum = e;
#pragma unroll
    for (int s = 16; s >= 1; s >>= 1)
      wsum += __shfl_xor(wsum, s, 32);
    if (lane == 0) s_wredB[wave] = wsum;
    __syncthreads();
    s_attn[h][m] = e / (s_wredB[h * 2] + s_wredB[h * 2 + 1]);
    __syncthreads();

    // prefetch next token's K tile while we chew through V
    if (t + 1 < TOK)
      __builtin_prefetch(K + (size_t)((tok + 1) * M_) * ADIM + tid * 32, 0, 3);

    // --- AV: 128 lanes <-> 128 contiguous channels (perfect coalescing),
    //     two halves of the m range in parallel ---
    {
      const int c    = tid & 127;
      const int half = tid >> 7;
      const int hc   = c >> 5;           // channel c = h*32 + d
      const float* vp = V + (size_t)(tok * M_) * ADIM + c;
      float acc = 0.f;
#pragma unroll 8
      for (int mm = 0; mm < 32; ++mm) {
        int mi = half * 32 + mm;
        acc += s_attn[hc][mi] *
               __builtin_nontemporal_load(vp + (size_t)mi * ADIM);  // TH=NT
      }
      s_red[tid] = acc;
    }
    __syncthreads();
    if (tid < ADIM) s_x[t][tid] += s_red[tid] + s_red[ADIM + tid];  // residual Q + attn
    __syncthreads();
  }

  // =================== x = LN1(Q + attn); y = LN2(x) ===================
  layernorm_tile(s_x, s_x, ln1g, ln1b, s_red, s_red2, s_mu, s_rs, tid);
  layernorm_tile(s_x, s_y, ln2g, ln2b, s_red, s_red2, s_mu, s_rs, tid);

  // =================== GEMM1: H = gelu(y @ W1 + b1) =====================
  // 16x512 = (16x128)@(128x512). 8 waves x 4 N-tiles each; one shared A-frag
  // per k-step, 4 accumulators to break the WMMA->WMMA RAW chain.
  {
    const int row  = lane & 15;          // A: M index
    const int ksel = (lane >> 4) << 1;   // A/B: lane-half selects K {0,1} vs {2,3}
    const int colb = lane & 15;          // B: N index within tile
    const int n0   = wave * 64;
    v8f acc0 = {}, acc1 = {}, acc2 = {}, acc3 = {};
#pragma unroll 4
    for (int k = 0; k < ADIM; k += 4) {
      v2f a; a.x = s_y[row][k + ksel]; a.y = s_y[row][k + ksel + 1];
      const float* wp = W1 + (size_t)(k + ksel) * HID + n0 + colb;
      v2f b0, b1v, b2v, b3v;
      b0.x  = wp[0];       b0.y  = wp[HID];
      b1v.x = wp[16];      b1v.y = wp[HID + 16];
      b2v.x = wp[32];      b2v.y = wp[HID + 32];
      b3v.x = wp[48];      b3v.y = wp[HID + 48];
      acc0 = wmma_f32_4(a, b0,  acc0);
      acc1 = wmma_f32_4(a, b1v, acc1);
      acc2 = wmma_f32_4(a, b2v, acc2);
      acc3 = wmma_f32_4(a, b3v, acc3);
    }
    // C layout: VGPR r -> M = r + 8*(lane>=16), N = lane&15
    const int mbase = (lane >> 4) << 3;
#pragma unroll
    for (int it = 0; it < 4; ++it) {
      const int nc = n0 + it * 16 + (lane & 15);
      const v8f* accp = (it == 0) ? &acc0 : (it == 1) ? &acc1 : (it == 2) ? &acc2 : &acc3;
      const float bb = bias1[nc];
#pragma unroll
      for (int r = 0; r < 8; ++r)
        s_h[mbase + r][nc] = gelu_exact((*accp)[r] + bb);
    }
  }
  __syncthreads();

  // ============ GEMM2: out = x + H @ W2 + b2, 8 waves x 1 N-tile ========
  {
    const int row  = lane & 15;
    const int ksel = (lane >> 4) << 1;
    const int colb = lane & 15;
    const int n0   = wave * 16;
    v8f accA = {}, accB = {};            // split-K to halve the RAW chain
#pragma unroll 4
    for (int k = 0; k < HID; k += 8) {
      {
        v2f a; a.x = s_h[row][k + ksel]; a.y = s_h[row][k + ksel + 1];
        const float* wp = W2 + (size_t)(k + ksel) * ADIM + n0 + colb;
        v2f b; b.x = wp[0]; b.y = wp[ADIM];
        accA = wmma_f32_4(a, b, accA);
      }
      {
        v2f a; a.x = s_h[row][k + 4 + ksel]; a.y = s_h[row][k + 4 + ksel + 1];
        const float* wp = W2 + (size_t)(k + 4 + ksel) * ADIM + n0 + colb;
        v2f b; b.x = wp[0]; b.y = wp[ADIM];
        accB = wmma_f32_4(a, b, accB);
      }
    }
    const int nc    = n0 + (lane & 15);
    const int mbase = (lane >> 4) << 3;
    const float bb  = bias2[nc];
#pragma unroll
    for (int r = 0; r < 8; ++r) {
      const int mrow = mbase + r;
      out[(size_t)(tokBase + mrow) * ADIM + nc] =
          s_x[mrow][nc] + accA[r] + accB[r] + bb;
    }
  }
}

extern "C" void kernel_launch(void* const* d_in, const int* in_sizes, int n_in,
                              void* d_out, int out_size, void* d_ws, size_t ws_size,
                              hipStream_t stream) {
  (void)in_sizes; (void)n_in; (void)out_size; (void)d_ws; (void)ws_size;
  const float*         Q    = (const float*)d_in[0];
  const float*         K    = (const float*)d_in[1];
  const float*         V    = (const float*)d_in[2];
  const unsigned char* mask = (const unsigned char*)d_in[3];
  const float*         ln1g = (const float*)d_in[4];
  const float*         ln1b = (const float*)d_in[5];
  const float*         ln2g = (const float*)d_in[6];
  const float*         ln2b = (const float*)d_in[7];
  const float*         W1   = (const float*)d_in[8];
  const float*         b1   = (const float*)d_in[9];
  const float*         W2   = (const float*)d_in[10];
  const float*         b2   = (const float*)d_in[11];
  float* out = (float*)d_out;

  dim3 grid((B_ * NR) / TOK);   // 2048 workgroups of 256 threads (8 wave32)
  bev_block_kernel<<<grid, TPB, 0, stream>>>(Q, K, V, mask, ln1g, ln1b,
                                             ln2g, ln2b, W1, b1, W2, b2, out);
}